// KnnRegressFromGED_40681930228157
// MI455X (gfx1250) — compile-verified
//
#include <hip/hip_runtime.h>
#include <math.h>
#include <stdint.h>

// KNN regression from a precomputed distance matrix.
//   ged : [NB_TEST, n_train] f32 distances (row-major)
//   y   : [n_train] f32 targets
//   out : [NB_TEST] f32 = sum(exp(-a*d_k)*y_k) / sum(exp(-a*d_k)) over 10 smallest d
//
// HBM-bound: 204.8 MB streamed once => ~8.8us floor at 23.3 TB/s.
// Streaming path: CDNA5 async global->LDS loads (ASYNCcnt), 4-deep pipeline.

#define KNN 10
#define WAVE 32
#define STAGE_F4 512          // float4 per stage  = 8 KB
#define STAGE_ELEMS 2048      // floats per stage
#define STAGE_BYTES 8192
#define J_PER_STAGE 16        // async b128 instructions per stage (512 B each)
#define NBUF 4                // 32 KB LDS, up to 3 stages in flight

__global__ __launch_bounds__(WAVE)
void knn_ged_kernel(const float* __restrict__ ged,
                    const float* __restrict__ y,
                    const float* __restrict__ coef,
                    float* __restrict__ out,
                    int n_train)
{
    __shared__ float4 sbuf[NBUF][STAGE_F4];   // 32 KB

    const int lane = (int)threadIdx.x;
    const int row  = (int)blockIdx.x;
    const float* rowp = ged + (size_t)row * (size_t)n_train;
    const unsigned long long gbase = (unsigned long long)(uintptr_t)rowp;
    // Low 32 bits of a generic pointer into LDS == LDS byte offset on CDNA5.
    const unsigned int lds0 = (unsigned int)(uintptr_t)(&sbuf[0][0]);

    // Per-lane sorted top-K (ascending): val[KNN-1] is the current threshold.
    float        val[KNN];
    unsigned int idx[KNN];
#pragma unroll
    for (int t = 0; t < KNN; ++t) { val[t] = 3.402823466e38f; idx[t] = 0xffffffffu; }

    auto consider = [&](float v, unsigned int i) {
        if (v < val[KNN - 1]) {
            float cv = v; unsigned int ci = i;
#pragma unroll
            for (int t = 0; t < KNN; ++t) {
                if (cv < val[t]) {
                    float tv = val[t]; val[t] = cv; cv = tv;
                    unsigned ti = idx[t]; idx[t] = ci; ci = ti;
                }
            }
        }
    };

    // Guarded quad: min-tree (3 v_min) + 1 compare on the common path.
    auto quad = [&](float4 v, unsigned int bi) {
        const float m = fminf(fminf(v.x, v.y), fminf(v.z, v.w));
        if (m < val[KNN - 1]) {
            consider(v.x, bi + 0u);
            consider(v.y, bi + 1u);
            consider(v.z, bi + 2u);
            consider(v.w, bi + 3u);
        }
    };

    const int nfull = n_train / STAGE_ELEMS;

    // Issue one stage (16 x b128 per wave = 8 KB) of async global->LDS copy.
    auto issue = [&](int s) {
        // Our earlier ds_load reads of this buffer must complete before the
        // async engine overwrites it.
        asm volatile("s_wait_dscnt 0x0" ::: "memory");
        unsigned int ldsv = lds0 + (unsigned)((s & (NBUF - 1)) * STAGE_BYTES + lane * 16);
        unsigned int voff = (unsigned)(s * STAGE_BYTES + lane * 16);
#pragma unroll
        for (int j = 0; j < J_PER_STAGE; ++j) {
            asm volatile("global_load_async_to_lds_b128 %0, %1, %2"
                         :: "v"(ldsv + (unsigned)(j * 512)),
                            "v"(voff + (unsigned)(j * 512)),
                            "s"(gbase)
                         : "memory");
        }
    };

    // ---- streamed scan with 4-deep async pipeline ----
    if (nfull > 0) {
        const int pre = nfull < 3 ? nfull : 3;
        for (int s = 0; s < pre; ++s) issue(s);

        for (int s = 0; s < nfull; ++s) {
            if (s + 3 < nfull) {
                issue(s + 3);
                asm volatile("s_wait_asynccnt 0x30" ::: "memory"); // <=48: stage s done
            } else {
                const int rem = nfull - 1 - s;                      // stages after s in flight
                if (rem >= 2)      asm volatile("s_wait_asynccnt 0x20" ::: "memory");
                else if (rem == 1) asm volatile("s_wait_asynccnt 0x10" ::: "memory");
                else               asm volatile("s_wait_asynccnt 0x0"  ::: "memory");
            }

            const float4* bp = &sbuf[s & (NBUF - 1)][0];
            const unsigned int ebase = (unsigned)s * STAGE_ELEMS + (unsigned)lane * 4u;
#pragma unroll 1
            for (int jb = 0; jb < J_PER_STAGE; jb += 4) {
                // batch 4 ds_load_b128, then filter
                const float4 a = bp[(jb + 0) * WAVE + lane];
                const float4 b = bp[(jb + 1) * WAVE + lane];
                const float4 c = bp[(jb + 2) * WAVE + lane];
                const float4 d = bp[(jb + 3) * WAVE + lane];
                const unsigned int bi = ebase + (unsigned)jb * 128u;
                quad(a, bi);
                quad(b, bi + 128u);
                quad(c, bi + 256u);
                quad(d, bi + 384u);
            }
        }
    }

    // ---- tail: direct global loads (scalar remainder non-temporal) ----
    const int t0 = nfull * STAGE_ELEMS;
#pragma unroll 1
    for (int i = t0 + lane * 4; i < n_train; i += WAVE * 4) {
        if (i + 3 < n_train) {
            quad(*(const float4*)(rowp + i), (unsigned)i);
        } else {
#pragma unroll 1
            for (int c = 0; c < 4; ++c)
                if (i + c < n_train)
                    consider(__builtin_nontemporal_load(rowp + i + c), (unsigned)(i + c));
        }
    }

    // ---- wave32 merge: 10 rounds of butterfly min-reduce, winner pops ----
    const float cd = coef[0];
    const float alpha = cd * cd;
    float num = 0.0f, den = 0.0f;

#pragma unroll 1
    for (int r = 0; r < KNN; ++r) {
        float        cv = val[0];
        unsigned int ci = idx[0];
#pragma unroll
        for (int p = 0; p < 5; ++p) {
            const int off = 16 >> p;
            const float        ov = __shfl_xor(cv, off, WAVE);
            const unsigned int oi = __shfl_xor(ci, off, WAVE);
            // total order (value, index) so every lane converges on same winner
            if (ov < cv || (ov == cv && oi < ci)) { cv = ov; ci = oi; }
        }
        // element indices are unique per lane list -> exactly one lane pops
        if (ci == idx[0] && cv == val[0]) {
#pragma unroll
            for (int t = 0; t < KNN - 1; ++t) { val[t] = val[t + 1]; idx[t] = idx[t + 1]; }
            val[KNN - 1] = 3.402823466e38f; idx[KNN - 1] = 0xffffffffu;
        }
        const float sim = __expf(-alpha * cv);
        num += sim * y[ci];   // same address across lanes -> broadcast load
        den += sim;
    }

    if (lane == 0) out[row] = num / den;
}

extern "C" void kernel_launch(void* const* d_in, const int* in_sizes, int n_in,
                              void* d_out, int out_size, void* d_ws, size_t ws_size,
                              hipStream_t stream)
{
    const float* ged  = (const float*)d_in[0];
    const float* yv   = (const float*)d_in[1];
    const float* coef = (const float*)d_in[2];
    float* out = (float*)d_out;

    const int nb_test = out_size;                 // 1024
    const int n_train = in_sizes[0] / nb_test;    // 50000

    knn_ged_kernel<<<dim3(nb_test), dim3(WAVE), 0, stream>>>(ged, yv, coef, out, n_train);
}